// TransmutationGNN_43181601194757
// MI455X (gfx1250) — compile-verified
//
#include <hip/hip_runtime.h>
#include <hip/hip_bf16.h>

typedef __attribute__((ext_vector_type(2))) float v2f;
typedef __attribute__((ext_vector_type(8))) float v8f;

#define NNODE_FEAT_IN  128
#define NNODE_FEAT_HID 64
#define NNODE_FEAT_OUT 32

// ---------------------------------------------------------------------------
// Degree kernels: deg starts at 1.0 (self-loop), count dst edges, then rsqrt.
// ---------------------------------------------------------------------------
__global__ void deg_init_kernel(float* deg, int N) {
    int i = blockIdx.x * blockDim.x + threadIdx.x;
    if (i < N) deg[i] = 1.0f;
}

__global__ void deg_count_kernel(float* deg, const int* __restrict__ dst, int E) {
    int e = blockIdx.x * blockDim.x + threadIdx.x;
    if (e < E) atomicAdd(&deg[dst[e]], 1.0f);
}

__global__ void deg_rsqrt_kernel(float* deg, int N) {
    int i = blockIdx.x * blockDim.x + threadIdx.x;
    if (i < N) deg[i] = rsqrtf(deg[i]);   // deg buffer now holds inv_sqrt
}

// ---------------------------------------------------------------------------
// WMMA fp32 GEMM: C[N x NC] = A[N x K] * W[K x NC]
// One wave computes a 16-row tile, NC/16 column tiles, K stepped by 4 using
// V_WMMA_F32_16X16X4_F32.
//
// A 16x4 layout : lanes 0-15 M=lane (K=k+0,k+1), lanes 16-31 M=lane-16 (K=k+2,k+3)
// B 4x16 layout : symmetric (N=lane&15, K pair by lane half)
// C/D 16x16 f32 : VGPR r -> M = r + 8*(lane>=16), N = lane&15
// ---------------------------------------------------------------------------
template<int K, int NC>
__global__ void gemm_wmma_kernel(const float* __restrict__ A,
                                 const float* __restrict__ W,
                                 float* __restrict__ C,
                                 int N) {
    constexpr int NT = NC / 16;
    const int wave = (int)((blockIdx.x * blockDim.x + threadIdx.x) >> 5);
    const int lane = threadIdx.x & 31;
    const int rowBase = wave * 16;
    if (rowBase >= N) return;                 // wave-uniform exit, EXEC stays full

    const int m  = lane & 15;                 // row-in-tile for A, col for B/C
    const int kb = (lane >> 4) << 1;          // 0 for lanes 0-15, 2 for lanes 16-31

    int arow = rowBase + m;
    if (arow >= N) arow = N - 1;              // clamp: duplicate row, store is guarded

    v8f acc[NT];
    #pragma unroll
    for (int c = 0; c < NT; ++c) acc[c] = (v8f){0.f,0.f,0.f,0.f,0.f,0.f,0.f,0.f};

    const float* arowp = A + (long long)arow * K;
    #pragma unroll 4
    for (int k = 0; k < K; k += 4) {
        v2f a;
        a.x = arowp[k + kb + 0];
        a.y = arowp[k + kb + 1];
        #pragma unroll
        for (int c = 0; c < NT; ++c) {
            const float* wp = W + (long long)(k + kb) * NC + c * 16 + m;
            v2f b;
            b.x = wp[0];
            b.y = wp[NC];
            acc[c] = __builtin_amdgcn_wmma_f32_16x16x4_f32(
                /*neg_a=*/false, a, /*neg_b=*/false, b,
                /*c_mod=*/(short)0, acc[c], /*reuse_a=*/false, /*reuse_b=*/false);
        }
    }

    const int mh = (lane >> 4) * 8;           // M offset of this lane half in C
    #pragma unroll
    for (int c = 0; c < NT; ++c) {
        #pragma unroll
        for (int r = 0; r < 8; ++r) {
            int row = rowBase + mh + r;
            if (row < N) C[(long long)row * NC + c * 16 + m] = acc[c][r];
        }
    }
}

// ---------------------------------------------------------------------------
// Aggregation init: self-loop contribution (+ optional bias).
// agg[i,f] = h[i,f] * inv[i]^2 (+ bias[f])
// ---------------------------------------------------------------------------
template<int D>
__global__ void agg_init_kernel(float* __restrict__ agg,
                                const float* __restrict__ h,
                                const float* __restrict__ inv,
                                const float* __restrict__ bias,
                                int N) {
    long long t = (long long)blockIdx.x * blockDim.x + threadIdx.x;
    long long total = (long long)N * D;
    if (t >= total) return;
    int i = (int)(t / D);
    int f = (int)(t % D);
    float s = inv[i];
    float v = h[t] * s * s;
    if (bias) v += bias[f];
    agg[t] = v;
}

// ---------------------------------------------------------------------------
// Edge scatter: D/4 threads per edge, each does a float4 gather + 4 atomics.
// Gather table (h) and scatter target (agg) both fit in the 192MB L2.
// ---------------------------------------------------------------------------
template<int D>
__global__ void scatter_kernel(float* __restrict__ agg,
                               const float* __restrict__ h,
                               const float* __restrict__ inv,
                               const int* __restrict__ src,
                               const int* __restrict__ dst,
                               int E) {
    constexpr int TPE = D / 4;
    long long t = (long long)blockIdx.x * blockDim.x + threadIdx.x;
    long long total = (long long)E * TPE;
    if (t >= total) return;
    int e = (int)(t / TPE);
    int f = (int)(t % TPE) * 4;
    int s = src[e];
    int d = dst[e];
    float nrm = inv[s] * inv[d];
    const float4 hv = *(const float4*)(h + (long long)s * D + f);
    float* ap = agg + (long long)d * D + f;
    atomicAdd(ap + 0, hv.x * nrm);
    atomicAdd(ap + 1, hv.y * nrm);
    atomicAdd(ap + 2, hv.z * nrm);
    atomicAdd(ap + 3, hv.w * nrm);
}

// ---------------------------------------------------------------------------
// Bias + ReLU for layer 1 output (after full aggregation).
// ---------------------------------------------------------------------------
template<int D>
__global__ void bias_relu_kernel(float* __restrict__ a,
                                 const float* __restrict__ bias,
                                 int N) {
    long long t = (long long)blockIdx.x * blockDim.x + threadIdx.x;
    long long total = (long long)N * D;
    if (t >= total) return;
    int f = (int)(t % D);
    float v = a[t] + bias[f];
    a[t] = v > 0.0f ? v : 0.0f;
}

// ---------------------------------------------------------------------------
// Host-side launcher.
// ---------------------------------------------------------------------------
static inline int gridFor(long long total, int block) {
    return (int)((total + block - 1) / block);
}

extern "C" void kernel_launch(void* const* d_in, const int* in_sizes, int n_in,
                              void* d_out, int out_size, void* d_ws, size_t ws_size,
                              hipStream_t stream) {
    const float* x  = (const float*)d_in[0];   // [N, 128]
    const int* edge = (const int*)d_in[1];     // [2, E] flat: row0=src, row1=dst
    const float* W1 = (const float*)d_in[2];   // [128, 64]
    const float* b1 = (const float*)d_in[3];   // [64]
    const float* W2 = (const float*)d_in[4];   // [64, 32]
    const float* b2 = (const float*)d_in[5];   // [32]
    float* out = (float*)d_out;                // [N, 32]

    const int N = in_sizes[0] / NNODE_FEAT_IN;
    const int E = in_sizes[1] / 2;
    const int* esrc = edge;
    const int* edst = edge + E;

    // Workspace layout (bytes), 256B-aligned slices.
    char* ws = (char*)d_ws;
    size_t off = 0;
    auto take = [&](size_t bytes) {
        char* p = ws + off;
        off += (bytes + 255) & ~(size_t)255;
        return p;
    };
    float* inv  = (float*)take((size_t)N * sizeof(float));                    // deg -> inv_sqrt
    float* h1   = (float*)take((size_t)N * NNODE_FEAT_HID * sizeof(float));   // x@W1
    float* agg1 = (float*)take((size_t)N * NNODE_FEAT_HID * sizeof(float));   // layer1 agg/relu
    float* h2   = (float*)take((size_t)N * NNODE_FEAT_OUT * sizeof(float));   // relu1@W2
    (void)ws_size;

    const int BLK = 256;

    // 1) degrees -> inv_sqrt
    deg_init_kernel<<<gridFor(N, BLK), BLK, 0, stream>>>(inv, N);
    deg_count_kernel<<<gridFor(E, BLK), BLK, 0, stream>>>(inv, edst, E);
    deg_rsqrt_kernel<<<gridFor(N, BLK), BLK, 0, stream>>>(inv, N);

    // 2) layer-1 GEMM: h1 = x @ W1   (fp32 WMMA, 4 waves/block, 16 rows/wave)
    {
        long long waves = (N + 15) / 16;
        int blocks = (int)((waves + 3) / 4);
        gemm_wmma_kernel<NNODE_FEAT_IN, NNODE_FEAT_HID>
            <<<blocks, 128, 0, stream>>>(x, W1, h1, N);
    }

    // 3) layer-1 aggregation: self-loops, edge scatter, bias+relu
    agg_init_kernel<NNODE_FEAT_HID>
        <<<gridFor((long long)N * NNODE_FEAT_HID, BLK), BLK, 0, stream>>>(
            agg1, h1, inv, nullptr, N);
    scatter_kernel<NNODE_FEAT_HID>
        <<<gridFor((long long)E * (NNODE_FEAT_HID / 4), BLK), BLK, 0, stream>>>(
            agg1, h1, inv, esrc, edst, E);
    bias_relu_kernel<NNODE_FEAT_HID>
        <<<gridFor((long long)N * NNODE_FEAT_HID, BLK), BLK, 0, stream>>>(agg1, b1, N);

    // 4) layer-2 GEMM: h2 = relu1 @ W2
    {
        long long waves = (N + 15) / 16;
        int blocks = (int)((waves + 3) / 4);
        gemm_wmma_kernel<NNODE_FEAT_HID, NNODE_FEAT_OUT>
            <<<blocks, 128, 0, stream>>>(agg1, W2, h2, N);
    }

    // 5) layer-2 aggregation straight into d_out (bias folded into init)
    agg_init_kernel<NNODE_FEAT_OUT>
        <<<gridFor((long long)N * NNODE_FEAT_OUT, BLK), BLK, 0, stream>>>(
            out, h2, inv, b2, N);
    scatter_kernel<NNODE_FEAT_OUT>
        <<<gridFor((long long)E * (NNODE_FEAT_OUT / 4), BLK), BLK, 0, stream>>>(
            out, h2, inv, esrc, edst, E);
}